// Improve_ff_block4_26637387169835
// MI455X (gfx1250) — compile-verified
//
#include <hip/hip_runtime.h>

typedef float v4f __attribute__((ext_vector_type(4)));
typedef float v2f __attribute__((ext_vector_type(2)));

__device__ __forceinline__ float vmax4(v4f v) {
    return fmaxf(fmaxf(v.x, v.y), fmaxf(v.z, v.w));
}

__device__ __forceinline__ v4f vmaxv(v4f a, v4f b) {
    v4f r;
    r.x = fmaxf(a.x, b.x);
    r.y = fmaxf(a.y, b.y);
    r.z = fmaxf(a.z, b.z);
    r.w = fmaxf(a.w, b.w);
    return r;
}

// ---------------------------------------------------------------------------
// Direct-load max-pool KxK (K = 2, 4, 8) on (N, C, 28K, 28K) -> (N, C, 28, 28).
// One thread per pooled output; NT b128/b64 loads (inputs are read exactly
// once, keep them out of the way of the L2-resident pooled maps).
// ---------------------------------------------------------------------------
template <int K>
__global__ void pool_kernel(const float* __restrict__ in,
                            float* __restrict__ out,
                            int total) {
    int idx = blockIdx.x * blockDim.x + threadIdx.x;
    if (idx >= total) return;

    const int W = 28 * K;
    int x     = idx % 28;
    int t     = idx / 28;
    int y     = t % 28;
    int plane = t / 28;

    const float* base = in + (size_t)plane * W * W
                           + (size_t)(y * K) * W
                           + (size_t)(x * K);

    __builtin_prefetch(base + (size_t)(K / 2) * W, 0, 0);

    if (K == 2) {
        v2f a = __builtin_nontemporal_load((const v2f*)base);
        v2f b = __builtin_nontemporal_load((const v2f*)(base + W));
        out[idx] = fmaxf(fmaxf(a.x, a.y), fmaxf(b.x, b.y));
        return;
    }

    v4f acc = __builtin_nontemporal_load((const v4f*)base);
#pragma unroll
    for (int dy = 0; dy < K; ++dy) {
        const float* row = base + (size_t)dy * W;
#pragma unroll
        for (int dx = 0; dx < K; dx += 4) {
            if (dy == 0 && dx == 0) continue;
            v4f v = __builtin_nontemporal_load((const v4f*)(row + dx));
            acc = vmaxv(acc, v);
        }
    }
    out[idx] = vmax4(acc);
}

// ---------------------------------------------------------------------------
// Async-pipelined 16x16 max-pool on (8, 16, 448, 448) -> (8, 16, 28, 28).
// Grid: (plane, quarter) = 8*16*4 = 512 blocks x 256 threads.
// Each block streams 7 "bands" (16 input rows = 28 KB, contiguous) through a
// 2-deep LDS double buffer using GLOBAL_LOAD_ASYNC_TO_LDS_B128 (ASYNCcnt),
// then reduces each band out of LDS: 8 lanes per output, 2 rows x 16 cols per
// lane (8 x ds_load_b128), wave32 __shfl_xor tree.
// ---------------------------------------------------------------------------
__global__ void pool16_async_kernel(const float* __restrict__ in,
                                    float* __restrict__ out) {
    constexpr int W    = 448;
    constexpr int BAND = 16 * W;             // 7168 floats = 28 KB per band
    __shared__ float buf[2 * BAND];          // 56 KB double buffer

    const int plane = blockIdx.x >> 2;       // n*16 + c
    const int q     = blockIdx.x & 3;        // quarter of the 28 bands
    const int b0    = q * 7;
    const int t     = threadIdx.x;

    const float* pbase = in + (size_t)plane * (W * W);

    // Issue one band's 28 KB as 7 per-lane async b128 copies into buf half.
    auto issue = [&](int band) {
        // ISA 10.2: flat shared address low 32 bits == wave LDS byte offset.
        unsigned ldsBase = (unsigned)(size_t)&buf[(band & 1) * BAND];
        unsigned long long g =
            (unsigned long long)(pbase + (size_t)band * BAND);
#pragma unroll
        for (int k = 0; k < 7; ++k) {
            unsigned           loff = ldsBase + (unsigned)((k * 256 + t) * 16);
            unsigned long long ga   = g + (unsigned long long)((k * 256 + t) * 16);
            asm volatile("global_load_async_to_lds_b128 %0, %1, off"
                         :: "v"(loff), "v"(ga)
                         : "memory");
        }
    };

    issue(b0);
    for (int b = b0; b < b0 + 7; ++b) {
        if (b + 1 < b0 + 7) {
            issue(b + 1);
            // Band b's 7 copies complete in order before band b+1's 7.
            asm volatile("s_wait_asynccnt 0x7" ::: "memory");
        } else {
            asm volatile("s_wait_asynccnt 0x0" ::: "memory");
        }
        __syncthreads();

        // Reduce band b: output row y=b, 28 outputs, 8 lanes per output.
        const float* bb   = &buf[(b & 1) * BAND];
        int x    = t >> 3;                   // 0..31 (only x<28 written)
        int part = t & 7;                    // 2 window rows per lane
        int xs   = x < 28 ? x : 27;          // clamp to stay inside buffer

        const float* r0 = bb + (size_t)(2 * part) * W + (size_t)xs * 16;
        const float* r1 = r0 + W;
        v4f acc = *(const v4f*)r0;
        acc = vmaxv(acc, *(const v4f*)(r0 + 4));
        acc = vmaxv(acc, *(const v4f*)(r0 + 8));
        acc = vmaxv(acc, *(const v4f*)(r0 + 12));
        acc = vmaxv(acc, *(const v4f*)(r1));
        acc = vmaxv(acc, *(const v4f*)(r1 + 4));
        acc = vmaxv(acc, *(const v4f*)(r1 + 8));
        acc = vmaxv(acc, *(const v4f*)(r1 + 12));
        float m = vmax4(acc);
        m = fmaxf(m, __shfl_xor(m, 1));
        m = fmaxf(m, __shfl_xor(m, 2));
        m = fmaxf(m, __shfl_xor(m, 4));
        if (part == 0 && x < 28)
            out[(size_t)plane * 784 + (size_t)b * 28 + x] = m;

        __syncthreads();  // everyone done reading buf[b&1] before reuse
    }
}

// ---------------------------------------------------------------------------
// out[n,c,y,x] = relu(p1[n,c&127] + p2[n,c&63] + p3[n,c&31] + p4[n,c&15]
//                     + ff[n,c]) over 28x28 planes. One thread per float4.
// ---------------------------------------------------------------------------
__global__ void fuse_kernel(const float* __restrict__ p1,
                            const float* __restrict__ p2,
                            const float* __restrict__ p3,
                            const float* __restrict__ p4,
                            const float* __restrict__ ff,
                            float* __restrict__ out,
                            int total4) {
    int i = blockIdx.x * blockDim.x + threadIdx.x;
    if (i >= total4) return;

    int flat  = i * 4;
    int r     = flat % 784;
    int plane = flat / 784;      // n*256 + c
    int c     = plane & 255;
    int n     = plane >> 8;

    size_t o1 = ((size_t)(n * 128 + (c & 127))) * 784 + r;
    size_t o2 = ((size_t)(n *  64 + (c &  63))) * 784 + r;
    size_t o3 = ((size_t)(n *  32 + (c &  31))) * 784 + r;
    size_t o4 = ((size_t)(n *  16 + (c &  15))) * 784 + r;

    v4f a = *(const v4f*)(p1 + o1);
    v4f b = *(const v4f*)(p2 + o2);
    v4f d = *(const v4f*)(p3 + o3);
    v4f e = *(const v4f*)(p4 + o4);
    v4f f = __builtin_nontemporal_load((const v4f*)(ff + flat));

    v4f s;
    s.x = fmaxf(a.x + b.x + d.x + e.x + f.x, 0.0f);
    s.y = fmaxf(a.y + b.y + d.y + e.y + f.y, 0.0f);
    s.z = fmaxf(a.z + b.z + d.z + e.z + f.z, 0.0f);
    s.w = fmaxf(a.w + b.w + d.w + e.w + f.w, 0.0f);

    __builtin_nontemporal_store(s, (v4f*)(out + flat));
}

extern "C" void kernel_launch(void* const* d_in, const int* in_sizes, int n_in,
                              void* d_out, int out_size, void* d_ws, size_t ws_size,
                              hipStream_t stream) {
    const float* in1 = (const float*)d_in[0];  // (8,128, 56, 56)
    const float* in2 = (const float*)d_in[1];  // (8, 64,112,112)
    const float* in3 = (const float*)d_in[2];  // (8, 32,224,224)
    const float* in4 = (const float*)d_in[3];  // (8, 16,448,448)
    const float* ff  = (const float*)d_in[4];  // (8,256, 28, 28)
    float* out = (float*)d_out;

    const int P1 = 8 * 128 * 28 * 28;  // 802816
    const int P2 = 8 *  64 * 28 * 28;  // 401408
    const int P3 = 8 *  32 * 28 * 28;  // 200704
    float* p1 = (float*)d_ws;
    float* p2 = p1 + P1;
    float* p3 = p2 + P2;
    float* p4 = p3 + P3;

    const int B = 256;  // 8 wave32 waves per block
    pool_kernel<2><<<(P1 + B - 1) / B, B, 0, stream>>>(in1, p1, P1);
    pool_kernel<4><<<(P2 + B - 1) / B, B, 0, stream>>>(in2, p2, P2);
    pool_kernel<8><<<(P3 + B - 1) / B, B, 0, stream>>>(in3, p3, P3);
    pool16_async_kernel<<<8 * 16 * 4, B, 0, stream>>>(in4, p4);

    const int T4 = 8 * 256 * 28 * 28 / 4;  // 401408 float4s
    fuse_kernel<<<(T4 + B - 1) / B, B, 0, stream>>>(p1, p2, p3, p4, ff, out, T4);
}